// AutomationGNN_1632087573166
// MI455X (gfx1250) — compile-verified
//
#include <hip/hip_runtime.h>

typedef __attribute__((ext_vector_type(2))) float v2f;
typedef __attribute__((ext_vector_type(8))) float v8f;

// ---------------------------------------------------------------------------
// Degree / normalization kernels
// ---------------------------------------------------------------------------
__global__ void deg_init(float* __restrict__ deg, int n) {
  int i = blockIdx.x * blockDim.x + threadIdx.x;
  if (i < n) deg[i] = 1.0f;  // self-loop
}

__global__ void deg_count(const long long* __restrict__ dst, float* __restrict__ deg,
                          long long E) {
  long long e = (long long)blockIdx.x * blockDim.x + threadIdx.x;
  if (e < E) unsafeAtomicAdd(&deg[dst[e]], 1.0f);
}

__global__ void deg_rsqrt(float* __restrict__ deg, int n) {
  int i = blockIdx.x * blockDim.x + threadIdx.x;
  if (i < n) deg[i] = rsqrtf(deg[i]);
}

// ---------------------------------------------------------------------------
// fp32 WMMA GEMM: H[N x OUTD] = X[N x K] * W[K x OUTD]
// One wave computes one 16x16 output tile, stepping K by 4
// (V_WMMA_F32_16X16X4_F32). K, OUTD compile-time -> all strides are
// immediate offsets; K-loop fully unrolled into a WMMA chain.
// N % 16 == 0 (50000 = 3125*16).
// ---------------------------------------------------------------------------
template <int K, int OUTD>
__global__ __launch_bounds__(128) void gcn_gemm_f32(const float* __restrict__ X,
                                                    const float* __restrict__ W,
                                                    float* __restrict__ H,
                                                    int totalWaves) {
  int wave = (blockIdx.x * blockDim.x + threadIdx.x) >> 5;
  if (wave >= totalWaves) return;  // wave-uniform
  int lane = threadIdx.x & 31;
  int lo = lane & 15;
  int hi = lane >> 4;  // 0: lanes 0-15, 1: lanes 16-31
  constexpr int colTiles = OUTD >> 4;  // power of two -> shift/mask
  int rowBase = (wave / colTiles) << 4;
  int colBase = (wave % colTiles) << 4;

  v8f c = {0.f, 0.f, 0.f, 0.f, 0.f, 0.f, 0.f, 0.f};
  // A: lane holds row M=lo, K pair starting at 2*hi
  const float* xrow = X + (size_t)(rowBase + lo) * K + 2 * hi;
  // B: lane holds col N=colBase+lo, K pair starting at 2*hi
  const float* wcol = W + (size_t)(2 * hi) * OUTD + colBase + lo;
#pragma unroll
  for (int k0 = 0; k0 < K; k0 += 4) {
    v2f a = *(const v2f*)(xrow + k0);            // 8B load, imm offset
    v2f b;
    b.x = wcol[(size_t)k0 * OUTD];               // imm offset (< 64KB)
    b.y = wcol[(size_t)(k0 + 1) * OUTD];
    c = __builtin_amdgcn_wmma_f32_16x16x4_f32(false, a, false, b, (short)0, c,
                                              false, false);
  }
  // D: VGPR v -> row M = v + 8*hi, col N = colBase + lo
  float* hrow = H + (size_t)(rowBase + 8 * hi) * OUTD + colBase + lo;
#pragma unroll
  for (int v = 0; v < 8; ++v) hrow[(size_t)v * OUTD] = c[v];
}

// ---------------------------------------------------------------------------
// A[i,c] = H[i,c] * dinv[i]^2   (self-loop contribution doubles as zero-init)
// ---------------------------------------------------------------------------
template <int F>
__global__ void selfloop_init(const float* __restrict__ H, const float* __restrict__ dinv,
                              float* __restrict__ A, long long total) {
  long long idx = (long long)blockIdx.x * blockDim.x + threadIdx.x;
  if (idx >= total) return;
  long long i = idx / F;  // F = power of two -> shift
  float di = dinv[i];
  A[idx] = H[idx] * di * di;
}

// ---------------------------------------------------------------------------
// Per-edge scatter: A[dst,:] += H[src,:] * dinv[src]*dinv[dst]
// One wave per edge. Edge indices forced scalar (readfirstlane) so the
// src/dst/dinv accesses become SALU/SMEM; per-lane work is one float4
// (or float2) gather + native f32 atomics.
// ---------------------------------------------------------------------------
template <int F>
__global__ __launch_bounds__(256) void edge_scatter(const float* __restrict__ H,
                                                    const float* __restrict__ dinv,
                                                    const long long* __restrict__ src,
                                                    const long long* __restrict__ dst,
                                                    float* __restrict__ A, int E) {
  int wave = (int)((blockIdx.x * (blockDim.x >> 5)) + (threadIdx.x >> 5));
  if (wave >= E) return;  // wave-uniform
  int lane = threadIdx.x & 31;
  int s = __builtin_amdgcn_readfirstlane((int)src[wave]);
  int d = __builtin_amdgcn_readfirstlane((int)dst[wave]);
  float norm = dinv[s] * dinv[d];
  constexpr int PER = F / 32;  // floats per lane: 4 (F=128) or 2 (F=64)
  const float* h = H + (size_t)s * F + lane * PER;
  float* o = A + (size_t)d * F + lane * PER;
  if constexpr (PER == 4) {
    float4 hv = *(const float4*)h;
    unsafeAtomicAdd(o + 0, hv.x * norm);
    unsafeAtomicAdd(o + 1, hv.y * norm);
    unsafeAtomicAdd(o + 2, hv.z * norm);
    unsafeAtomicAdd(o + 3, hv.w * norm);
  } else {
    float2 hv = *(const float2*)h;
    unsafeAtomicAdd(o + 0, hv.x * norm);
    unsafeAtomicAdd(o + 1, hv.y * norm);
  }
}

// ---------------------------------------------------------------------------
// Xout[i,c] = relu(A[i,c] + b[c])  /  bias-only (in place) for the last layer
// ---------------------------------------------------------------------------
template <int F>
__global__ void bias_relu(const float* __restrict__ A, const float* __restrict__ b,
                          float* __restrict__ X, long long total) {
  long long idx = (long long)blockIdx.x * blockDim.x + threadIdx.x;
  if (idx >= total) return;
  int c = (int)(idx & (F - 1));
  float v = A[idx] + b[c];
  X[idx] = v > 0.f ? v : 0.f;
}

template <int F>
__global__ void bias_add_inplace(float* __restrict__ A, const float* __restrict__ b,
                                 long long total) {
  long long idx = (long long)blockIdx.x * blockDim.x + threadIdx.x;
  if (idx >= total) return;
  int c = (int)(idx & (F - 1));
  A[idx] += b[c];
}

// ---------------------------------------------------------------------------
static inline unsigned cdiv_ll(long long a, long long b) {
  return (unsigned)((a + b - 1) / b);
}

extern "C" void kernel_launch(void* const* d_in, const int* in_sizes, int n_in,
                              void* d_out, int out_size, void* d_ws, size_t ws_size,
                              hipStream_t stream) {
  const float* x  = (const float*)d_in[0];
  const long long* ei = (const long long*)d_in[1];  // int64 [2, E]
  const float* W1 = (const float*)d_in[2];
  const float* b1 = (const float*)d_in[3];
  const float* W2 = (const float*)d_in[4];
  const float* b2 = (const float*)d_in[5];
  const float* W3 = (const float*)d_in[6];
  const float* b3 = (const float*)d_in[7];

  const int IN_D = 128, HID = 128, OUT_D = 64;
  const int N = in_sizes[0] / IN_D;        // 50000
  const long long E = in_sizes[1] / 2;     // 800000
  const long long* src = ei;
  const long long* dst = ei + E;

  // Workspace layout (fp32): dinv | H | A | Xf
  float* ws   = (float*)d_ws;
  float* dinv = ws;
  size_t npad = ((size_t)N + 63) & ~(size_t)63;
  float* H  = dinv + npad;
  float* A  = H + (size_t)N * HID;
  float* Xf = A + (size_t)N * HID;

  const int TB = 256;
  const int wavesPerBlk = TB / 32;
  const unsigned edgeBlocks = cdiv_ll(E, wavesPerBlk);

  // ---- degrees -> dinv ----
  deg_init<<<cdiv_ll(N, TB), TB, 0, stream>>>(dinv, N);
  deg_count<<<cdiv_ll(E, TB), TB, 0, stream>>>(dst, dinv, E);
  deg_rsqrt<<<cdiv_ll(N, TB), TB, 0, stream>>>(dinv, N);

  const long long tot128 = (long long)N * HID;
  const long long tot64  = (long long)N * OUT_D;

  // ---- layer 1: x(128) -> Xf(128), relu ----
  {
    int waves = (N / 16) * (HID / 16);
    gcn_gemm_f32<128, 128><<<cdiv_ll((long long)waves * 32, 128), 128, 0, stream>>>(
        x, W1, H, waves);
    selfloop_init<128><<<cdiv_ll(tot128, TB), TB, 0, stream>>>(H, dinv, A, tot128);
    edge_scatter<128><<<edgeBlocks, TB, 0, stream>>>(H, dinv, src, dst, A, (int)E);
    bias_relu<128><<<cdiv_ll(tot128, TB), TB, 0, stream>>>(A, b1, Xf, tot128);
  }

  // ---- layer 2: Xf(128) -> Xf(128), relu ----
  {
    int waves = (N / 16) * (HID / 16);
    gcn_gemm_f32<128, 128><<<cdiv_ll((long long)waves * 32, 128), 128, 0, stream>>>(
        Xf, W2, H, waves);
    selfloop_init<128><<<cdiv_ll(tot128, TB), TB, 0, stream>>>(H, dinv, A, tot128);
    edge_scatter<128><<<edgeBlocks, TB, 0, stream>>>(H, dinv, src, dst, A, (int)E);
    bias_relu<128><<<cdiv_ll(tot128, TB), TB, 0, stream>>>(A, b2, Xf, tot128);
  }

  // ---- layer 3: Xf(128) -> out(64), no relu ----
  {
    float* out = (float*)d_out;
    int waves = (N / 16) * (OUT_D / 16);
    gcn_gemm_f32<128, 64><<<cdiv_ll((long long)waves * 32, 128), 128, 0, stream>>>(
        Xf, W3, H, waves);
    selfloop_init<64><<<cdiv_ll(tot64, TB), TB, 0, stream>>>(H, dinv, out, tot64);
    edge_scatter<64><<<edgeBlocks, TB, 0, stream>>>(H, dinv, src, dst, out, (int)E);
    bias_add_inplace<64><<<cdiv_ll(tot64, TB), TB, 0, stream>>>(out, b3, tot64);
  }
}